// rbfLayer_687194767964
// MI455X (gfx1250) — compile-verified
//
#include <hip/hip_runtime.h>
#include <hip/hip_bf16.h>

// RBF continuous-conv layer for MI455X (gfx1250), wave32.
//
//   out[p,i] = sum_k W[i,k] * vsum[p,k],  k = j*16 + n*4 + m  (K = 128)
//   W[i,k]    = kernel[i,j,n,m]
//   vsum[p,k] = sum_{e in p} features[nbr(e), j] * c[e,n]*f[e,m]
//
// GEMM tile per wave: D(16x16) = A(16x4) x B(4x16) chained over K=128 with
// V_WMMA_F32_16X16X4_F32 (f32 exact vs reference).
//   A rows  = output features (8 real + 8 zero-pad)
//   B cols  = 16 points owned by this wave
// Per ISA 7.12.2 (32-bit A 16x4): lane m-index = lane&15, K pair = {2*half, 2*half+1}
// within each K-step of 4 (half = lane>>4). B assumed mirrored (N across lanes).

typedef __attribute__((ext_vector_type(2))) float v2f;
typedef __attribute__((ext_vector_type(8))) float v8f;

#define NB   4
#define MB   4
#define FIN  8
#define FOUT 8
#define DEG  16
#define KTOT 128                        // FIN * NB * MB
#define NSTEP 32                        // KTOT / 4 per-WMMA K

#define WAVES_PER_BLOCK 2
#define PTS_PER_WAVE    16
#define EDGES_PER_TILE  (PTS_PER_WAVE * DEG)   // 256
#define S_STRIDE  17                    // 16 basis vals + pad (odd -> bank friendly)
#define FV_STRIDE 9                     // 8 feats + pad

__global__ __launch_bounds__(WAVES_PER_BLOCK * 32, 1)
void rbfLayer_687194767964_kernel(const float* __restrict__ pos,    // [N,2]
                                  const float* __restrict__ feat,   // [N,FIN]
                                  const float* __restrict__ kern,   // [FOUT,FIN,NB,MB]
                                  const int*   __restrict__ nbrs,   // [E]
                                  float* __restrict__ out,          // [N,FOUT]
                                  int n_pts) {
    __shared__ float sBuf[WAVES_PER_BLOCK][EDGES_PER_TILE * S_STRIDE];
    __shared__ float fvBuf[WAVES_PER_BLOCK][EDGES_PER_TILE * FV_STRIDE];

    const int lane = threadIdx.x & 31;
    const int wave = threadIdx.x >> 5;
    const int half = lane >> 4;          // selects K pair within each 4-wide K step
    const int lp   = lane & 15;          // A: out-feature row / B,D: point column
    const int tileBase = ((int)blockIdx.x * WAVES_PER_BLOCK + wave) * PTS_PER_WAVE;

    float* sW = sBuf[wave];
    float* fW = fvBuf[wave];

    // ---- A fragments: weights, resident in VGPRs for the whole kernel ----
    float aW[NSTEP][2];
#pragma unroll
    for (int t = 0; t < NSTEP; ++t) {
#pragma unroll
        for (int q = 0; q < 2; ++q) {
            int k  = 4 * t + 2 * half + q;       // global K index
            int j  = k >> 4;
            int nm = k & 15;
            aW[t][q] = (lp < FOUT) ? kern[(lp * FIN + j) * (NB * MB) + nm] : 0.0f;
        }
    }

    // ---- Phase 1: per-edge geometry + bases + feature gather -> LDS ----
#pragma unroll
    for (int u = 0; u < EDGES_PER_TILE / 32; ++u) {   // 8 edges per lane
        int eLocal = lane * (EDGES_PER_TILE / 32) + u; // 0..255
        int pLoc   = eLocal >> 4;                      // owning point in tile
        int eIn    = eLocal & (DEG - 1);               // edge within point
        int pGlob  = tileBase + pLoc;
        int slot   = eIn * PTS_PER_WAVE + pLoc;        // point-minor -> odd-stride LDS reads

        float s[NB * MB];
        float fv[FIN];
        if (pGlob < n_pts) {
            int eGlob = pGlob * DEG + eIn;
            int nb    = nbrs[eGlob];
            float ox = pos[2 * pGlob], oy = pos[2 * pGlob + 1];
            float nx = pos[2 * nb],    ny = pos[2 * nb + 1];
            float rx = nx - ox, ry = ny - oy;
            float r  = sqrtf(rx * rx + ry * ry) * 0.5f;          // / SUPPORT(=2)
            float xr = r * 2.0f - 1.0f;
            float xt = atan2f(ry, rx) * 0.31830988618379067f;    // theta / pi

            float c[NB], f[MB];
#pragma unroll
            for (int n = 0; n < NB; ++n) {                       // non-periodic hats
                float cent = -1.0f + (float)n * (2.0f / 3.0f);   // spacing 2/(NB-1)
                float dd = fabsf(xr - cent) * 1.5f;
                c[n] = fminf(fmaxf(1.0f - dd, 0.0f), 1.0f);
            }
#pragma unroll
            for (int m = 0; m < MB; ++m) {                       // periodic hats
                float cent = -1.0f + (float)m * 0.5f;            // spacing 2/MB
                float ra = xt - cent;
                float dd = fminf(fminf(fabsf(ra), fabsf(ra - 2.0f)), fabsf(ra + 2.0f)) * 2.0f;
                f[m] = fminf(fmaxf(1.0f - dd, 0.0f), 1.0f);
            }
#pragma unroll
            for (int n = 0; n < NB; ++n)
#pragma unroll
                for (int m = 0; m < MB; ++m)
                    s[n * MB + m] = c[n] * f[m];
#pragma unroll
            for (int j = 0; j < FIN; ++j)
                fv[j] = feat[nb * FIN + j];
        } else {
#pragma unroll
            for (int q = 0; q < NB * MB; ++q) s[q] = 0.0f;
#pragma unroll
            for (int j = 0; j < FIN; ++j) fv[j] = 0.0f;
        }
#pragma unroll
        for (int q = 0; q < NB * MB; ++q) sW[slot * S_STRIDE + q] = s[q];
#pragma unroll
        for (int j = 0; j < FIN; ++j)     fW[slot * FV_STRIDE + j] = fv[j];
    }

    __syncthreads();

    // ---- Phase 2: accumulate this lane's B-fragment slice of vsum ----
    // vacc[t][q] = vsum[point=lp][k = 4t + 2*half + q]
    float vacc[NSTEP][2];
#pragma unroll
    for (int t = 0; t < NSTEP; ++t) { vacc[t][0] = 0.0f; vacc[t][1] = 0.0f; }

#pragma unroll 4
    for (int e = 0; e < DEG; ++e) {
        int slot = e * PTS_PER_WAVE + lp;
        const float* sr = &sW[slot * S_STRIDE];
        const float* fr = &fW[slot * FV_STRIDE];
        float sv[8];
#pragma unroll
        for (int q = 0; q < 4; ++q) {            // nm = 4q + 2*half {,+1}
            sv[2 * q]     = sr[4 * q + 2 * half];
            sv[2 * q + 1] = sr[4 * q + 2 * half + 1];
        }
#pragma unroll
        for (int j = 0; j < FIN; ++j) {
            float fj = fr[j];
#pragma unroll
            for (int q = 0; q < 4; ++q) {
                int t = j * 4 + q;               // k = 16j + 4q + 2*half (+1)
                vacc[t][0] = fmaf(fj, sv[2 * q],     vacc[t][0]);
                vacc[t][1] = fmaf(fj, sv[2 * q + 1], vacc[t][1]);
            }
        }
    }

    // ---- WMMA chain: D = sum_t A_t x B_t  (f32 16x16x4, K = 128) ----
    v8f d = {};
#pragma unroll
    for (int t = 0; t < NSTEP; ++t) {
        v2f a; a.x = aW[t][0];   a.y = aW[t][1];
        v2f b; b.x = vacc[t][0]; b.y = vacc[t][1];
        d = __builtin_amdgcn_wmma_f32_16x16x4_f32(
                /*neg_a=*/false, a, /*neg_b=*/false, b,
                /*c_mod=*/(short)0, d, /*reuse_a=*/false, /*reuse_b=*/false);
    }

    // ---- Store: lane r<16 holds point (tileBase+lp), features d[0..7] ----
    int pGlob = tileBase + lp;
    if (half == 0 && pGlob < n_pts) {
        float4 lo = make_float4(d[0], d[1], d[2], d[3]);
        float4 hi = make_float4(d[4], d[5], d[6], d[7]);
        float* op = out + (size_t)pGlob * FOUT;
        *(float4*)(op)     = lo;
        *(float4*)(op + 4) = hi;
    }
}

extern "C" void kernel_launch(void* const* d_in, const int* in_sizes, int n_in,
                              void* d_out, int out_size, void* d_ws, size_t ws_size,
                              hipStream_t stream) {
    (void)n_in; (void)out_size; (void)d_ws; (void)ws_size;
    const float* pos  = (const float*)d_in[0];   // [N,2]
    const float* feat = (const float*)d_in[1];   // [N,FIN]
    const float* kern = (const float*)d_in[2];   // [FOUT,FIN,NB,MB]
    const int*   nbrs = (const int*)d_in[3];     // [E]
    // d_in[4] = row_splits: uniform (p*DEG), folded into indexing.

    int n_pts = in_sizes[0] / 2;
    int ptsPerBlock = WAVES_PER_BLOCK * PTS_PER_WAVE;       // 32
    int grid = (n_pts + ptsPerBlock - 1) / ptsPerBlock;

    rbfLayer_687194767964_kernel<<<grid, WAVES_PER_BLOCK * 32, 0, stream>>>(
        pos, feat, kern, nbrs, (float*)d_out, n_pts);
}